// Model_75428215653240
// MI455X (gfx1250) — compile-verified
//
#include <hip/hip_runtime.h>

// Reference shapes (compile-time constants from the Python reference):
//   img:       [64, 3, 416, 416] f32
//   image_ref: [ 1, 3, 608, 608] f32
//   out:       [64, 3, 608, 608] f32
//   paste offsets: ROW_OFF=85, COL_OFF=80
//
// Pure bandwidth-bound composite (~420 MB HBM traffic -> ~18 us at 23.3 TB/s).
// Round-2 change: map (y,b) onto (blockIdx.x, blockIdx.y) so row/batch
// indices are SGPR-uniform -> no vector integer division chains, row-region
// test becomes a scalar branch, and addressing folds into saddr+lane-offset.
// Access pattern unchanged: 9 coalesced b128 ops per (b,y,x4) site, with
// NT hints on the read-once img stream and write-once out stream so the
// 4.4 MB image_ref stays resident in the 192 MB L2 across all 64 batches.

typedef float v4f __attribute__((ext_vector_type(4)));

namespace {
constexpr int Bn      = 64;
constexpr int Hh      = 416, Ww = 416;   // img spatial dims
constexpr int H       = 608, W  = 608;   // ref/out spatial dims
constexpr int ROW_OFF = 85,  COL_OFF = 80;

constexpr int W4     = W / 4;            // 152 float4 per out row
constexpr int w4     = Ww / 4;           // 104 float4 per img row
constexpr int COL4   = COL_OFF / 4;      // 20  (region is float4-aligned)
constexpr int REF_CH = H * W4;           // ref/out channel stride (float4)
constexpr int IMG_CH = Hh * w4;          // img channel stride (float4)

constexpr int THREADS = 160;             // 5 wave32; covers 152 cols, 5% idle
}

__global__ __launch_bounds__(THREADS) void composite_kernel(
    const v4f* __restrict__ img,
    const v4f* __restrict__ ref,
    v4f* __restrict__ out) {
  const int x4 = threadIdx.x;            // float4 column, lane-varying
  const int y  = blockIdx.x;             // row,   SGPR-uniform
  const int b  = blockIdx.y;             // batch, SGPR-uniform
  if (x4 >= W4) return;

  // image_ref reads: regular-temporal -> L2-resident across all 64 batches.
  const v4f* refp = ref + (long long)y * W4 + x4;
  const v4f r0 = refp[0];
  const v4f r1 = refp[REF_CH];
  const v4f r2 = refp[2 * REF_CH];

  v4f o0 = r0, o1 = r1, o2 = r2;

  const int iy = y - ROW_OFF;            // scalar: whole block in/out of region
  const int ix = x4 - COL4;              // lane-varying column test
  if ((unsigned)iy < (unsigned)Hh && (unsigned)ix < (unsigned)w4) {
    // Streaming read-once data: non-temporal so it doesn't evict image_ref.
    const v4f* ip = img + (long long)b * (3 * IMG_CH) + (long long)iy * w4 + ix;
    const v4f i0 = __builtin_nontemporal_load(&ip[0]);
    const v4f i1 = __builtin_nontemporal_load(&ip[IMG_CH]);
    const v4f m  = __builtin_nontemporal_load(&ip[2 * IMG_CH]);  // mask == channel-2 data

    // Branchless per-component select (v_cmp + v_cndmask).
    o0.x = (m.x != 0.0f) ? i0.x : r0.x;
    o0.y = (m.y != 0.0f) ? i0.y : r0.y;
    o0.z = (m.z != 0.0f) ? i0.z : r0.z;
    o0.w = (m.w != 0.0f) ? i0.w : r0.w;

    o1.x = (m.x != 0.0f) ? i1.x : r1.x;
    o1.y = (m.y != 0.0f) ? i1.y : r1.y;
    o1.z = (m.z != 0.0f) ? i1.z : r1.z;
    o1.w = (m.w != 0.0f) ? i1.w : r1.w;

    o2.x = (m.x != 0.0f) ? m.x : r2.x;
    o2.y = (m.y != 0.0f) ? m.y : r2.y;
    o2.z = (m.z != 0.0f) ? m.z : r2.z;
    o2.w = (m.w != 0.0f) ? m.w : r2.w;
  }

  // Write-once output stream: non-temporal b128 stores.
  v4f* op = out + (long long)b * (3 * REF_CH) + (long long)y * W4 + x4;
  __builtin_nontemporal_store(o0, &op[0]);
  __builtin_nontemporal_store(o1, &op[REF_CH]);
  __builtin_nontemporal_store(o2, &op[2 * REF_CH]);
}

extern "C" void kernel_launch(void* const* d_in, const int* in_sizes, int n_in,
                              void* d_out, int out_size, void* d_ws, size_t ws_size,
                              hipStream_t stream) {
  (void)in_sizes; (void)n_in; (void)d_ws; (void)ws_size; (void)out_size;

  const v4f* img = (const v4f*)d_in[0];   // 64*3*416*416 f32 (multiple of 4)
  const v4f* ref = (const v4f*)d_in[1];   // 1*3*608*608  f32 (multiple of 4)
  v4f*       out = (v4f*)d_out;           // 64*3*608*608 f32

  dim3 grid(H, Bn, 1);                    // (608 rows, 64 batches)
  composite_kernel<<<grid, THREADS, 0, stream>>>(img, ref, out);
}